// DynamicWeights_1580547974800
// MI455X (gfx1250) — compile-verified
//
#include <hip/hip_runtime.h>
#include <math.h>

typedef __attribute__((ext_vector_type(2))) float v2f;
typedef __attribute__((ext_vector_type(8))) float v8f;

#define HW 192
#define CH 64
#define TILE 16
#define LDSROW 18
#define PSTRIDE 672   // channel-pair stride in floats: 324*2 padded so PSTRIDE % 64 == 32

__global__ __launch_bounds__(256) void dynconv_wmma_kernel(
    const float* __restrict__ x,       // (8,64,192,192)
    const float* __restrict__ conv_w,  // (9,64,3,3) OIHW
    const float* __restrict__ gamma,   // (1,)
    float* __restrict__ out)           // (8,64,192,192)
{
    // x tile with halo, channel-pair interleaved: [cp(32)][spatial(324)][2]  (86016 B)
    __shared__ __align__(16) float x_lds[32 * PSTRIDE];
    // B panel, K-pair interleaved: [kp(288)][col(16)][2], cols 9..15 zero    (36864 B)
    __shared__ __align__(16) float w2[288 * 32];
    // per-pixel 9 tap logits -> softmax weights                               (9216 B)
    __shared__ float filt[256 * 9];

    const int tid = threadIdx.x;
    const int wg  = blockIdx.x;
    const int n    = wg / 144;
    const int t144 = wg - n * 144;
    const int ty   = t144 / 12;
    const int tx   = t144 - ty * 12;
    const int h0   = ty * TILE;
    const int w0   = tx * TILE;

    // ---- stage haloed x tile into LDS, 2 channels per b64 store ----
    const float* xn = x + (size_t)n * CH * HW * HW;
    for (int idx = tid; idx < 32 * 324; idx += 256) {
        int cp  = idx / 324;
        int sp  = idx - cp * 324;
        int r   = sp / 18;
        int wl  = sp - r * 18;
        int hg  = h0 - 1 + r;
        int wg2 = w0 - 1 + wl;
        v2f v = {0.0f, 0.0f};
        if (hg >= 0 && hg < HW && wg2 >= 0 && wg2 < HW) {
            const float* p = xn + (size_t)(2 * cp) * (HW * HW) + hg * HW + wg2;
            v[0] = p[0];
            v[1] = p[HW * HW];
        }
        *(v2f*)&x_lds[cp * PSTRIDE + sp * 2] = v;
    }
    // ---- stage reorganized weight panel: w2[kp*32 + o*2 + par], k = tap*64 + c ----
    for (int idx = tid; idx < 288 * 32; idx += 256) {
        int kp  = idx >> 5;
        int rem = idx & 31;
        int o   = rem >> 1;
        int par = rem & 1;
        int k   = 2 * kp + par;
        int tap = k >> 6;
        int c   = k & 63;
        w2[idx] = (o < 9) ? conv_w[(o * CH + c) * 9 + tap] : 0.0f;
    }
    __syncthreads();

    // ---- conv logits via V_WMMA_F32_16X16X4_F32 ----
    // Each wave owns two 16-pixel tile rows (g0, g1). M = pixel col, N = tap, K = tap*64+c.
    const int lane = tid & 31;
    const int wv   = tid >> 5;
    const int col  = lane & 15;   // A: M ; B: N
    const int ksel = lane >> 4;   // lanes 16..31 carry the K+2 pair (32-bit 16x4 layout)
    const int g0   = wv * 2;
    const int g1   = g0 + 1;

    v8f acc0 = {};
    v8f acc1 = {};
    for (int tap = 0; tap < 9; ++tap) {
        const int i = tap / 3;
        const int j = tap - 3 * i;
        const int sp0 = ((g0 + i) * LDSROW + col + j) * 2;
        const int sp1 = sp0 + 2 * LDSROW;
#pragma unroll
        for (int q = 0; q < 16; ++q) {
            const int cp = 2 * q + ksel;               // channel pair for this lane's K pair
            v2f bfrag = *(const v2f*)&w2[(tap * 32 + cp) * 32 + col * 2];
            v2f a0    = *(const v2f*)&x_lds[cp * PSTRIDE + sp0];
            acc0 = __builtin_amdgcn_wmma_f32_16x16x4_f32(
                false, a0, false, bfrag, (short)0, acc0, false, false);
            v2f a1    = *(const v2f*)&x_lds[cp * PSTRIDE + sp1];
            acc1 = __builtin_amdgcn_wmma_f32_16x16x4_f32(
                false, a1, false, bfrag, (short)0, acc1, false, false);
        }
    }

    // ---- scatter C tiles to LDS: vgpr r, lane L holds C[M=r+8*(L>>4)][N=L&15] ----
    if (col < 9) {
#pragma unroll
        for (int r = 0; r < 8; ++r) {
            const int M0 = r + 8 * ksel;
            filt[((g0 << 4) + M0) * 9 + col] = acc0[r];
            filt[((g1 << 4) + M0) * 9 + col] = acc1[r];
        }
    }
    __syncthreads();

    // ---- softmax over the 9 taps, one pixel per thread ----
    {
        float l[9];
        const int base = tid * 9;
        float m = -INFINITY;
#pragma unroll
        for (int o = 0; o < 9; ++o) { l[o] = filt[base + o]; m = fmaxf(m, l[o]); }
        float s = 0.0f;
#pragma unroll
        for (int o = 0; o < 9; ++o) { l[o] = expf(l[o] - m); s += l[o]; }
        const float inv = 1.0f / s;
#pragma unroll
        for (int o = 0; o < 9; ++o) filt[base + o] = l[o] * inv;
    }
    __syncthreads();

    // ---- weighted 3x3 gather (2 channels at a time) + residual, all reads from LDS ----
    {
        const float gm = gamma[0];
        float f[9];
        const int base = tid * 9;
#pragma unroll
        for (int o = 0; o < 9; ++o) f[o] = filt[base + o];
        const int g = tid >> 4;
        const int m = tid & 15;
        const int h = h0 + g;
        const int w = w0 + m;
        float* outp = out + (size_t)n * CH * HW * HW + (size_t)h * HW + w;
        const int sp = (g * LDSROW + m) * 2;   // tap(i,j) at +(i*18+j)*2; center at +38
#pragma unroll 4
        for (int cp = 0; cp < 32; ++cp) {
            const float* xc = &x_lds[cp * PSTRIDE + sp];
            v2f t0 = *(const v2f*)&xc[0];
            v2f t1 = *(const v2f*)&xc[2];
            v2f t2 = *(const v2f*)&xc[4];
            v2f t3 = *(const v2f*)&xc[36];
            v2f t4 = *(const v2f*)&xc[38];
            v2f t5 = *(const v2f*)&xc[40];
            v2f t6 = *(const v2f*)&xc[72];
            v2f t7 = *(const v2f*)&xc[74];
            v2f t8 = *(const v2f*)&xc[76];
            v2f acc = t0 * f[0] + t1 * f[1] + t2 * f[2]
                    + t3 * f[3] + t4 * f[4] + t5 * f[5]
                    + t6 * f[6] + t7 * f[7] + t8 * f[8];
            outp[(size_t)(2 * cp)     * (HW * HW)] = fmaf(gm, acc[0], t4[0]);
            outp[(size_t)(2 * cp + 1) * (HW * HW)] = fmaf(gm, acc[1], t4[1]);
        }
    }
}

extern "C" void kernel_launch(void* const* d_in, const int* in_sizes, int n_in,
                              void* d_out, int out_size, void* d_ws, size_t ws_size,
                              hipStream_t stream) {
    const float* x      = (const float*)d_in[0];
    const float* conv_w = (const float*)d_in[1];
    const float* gamma  = (const float*)d_in[2];
    float* out          = (float*)d_out;
    // 8 images x 12x12 tiles of 16x16 pixels
    dynconv_wmma_kernel<<<dim3(8 * 144), dim3(256), 0, stream>>>(x, conv_w, gamma, out);
}